// MultiAttentionLoss_12713103196633
// MI455X (gfx1250) — compile-verified
//
#include <hip/hip_runtime.h>

#define NEG_FILL (-999999.0f)

typedef __attribute__((ext_vector_type(2))) float v2f;
typedef __attribute__((ext_vector_type(8))) float v8f;

#if defined(__has_builtin)
#if __has_builtin(__builtin_amdgcn_wmma_f32_16x16x4_f32)
#define HAVE_WMMA_F32X4 1
#endif
#endif

// ---------------------------------------------------------------------------
// Kernel A: dis_loss partials. One block = 16 maps (16 x 784 f32 staged in LDS).
// Per map: argmax (first-index tie-break) + 5 weighted sums via WMMA f32 16x16x4:
//   D(16 maps x 16 cols) += A(16x4 chunk of M) x B(4x16 basis [1,y,x,y^2,x^2,0..])
// ---------------------------------------------------------------------------
__global__ __launch_bounds__(256) void dis_kernel(const float* __restrict__ M,
                                                  float* __restrict__ wsDis) {
    const int HW = 784;
    const int STRIDE = 788;  // padded row stride (words): conflict-free WMMA A loads

    __shared__ __align__(16) float tile[16 * 788];   // 50432 B
#ifdef HAVE_WMMA_F32X4
    __shared__ float sumsA[8 * 80];                  // per-wave D dumps (cols 0..4)
#else
    __shared__ float thrS[256 * 5];                  // per-thread partial sums
#endif
    __shared__ float sums[80];                       // combined [map][5]
    __shared__ float amaxV[256];
    __shared__ int   amaxI[256];
    __shared__ int   mapIdx[16];
    __shared__ float contrib[16];

    const int tid = threadIdx.x;
    const int g = blockIdx.x;
    const long long base = (long long)g * 16 * HW;

    // ---- phase 1: coalesced float4 load of 16x784 into padded LDS tile ----
    const float4* g4 = (const float4*)(M + base);
    for (int i = tid; i < (16 * HW) / 4; i += 256) {
        int idx = i * 4;
        int row = idx / HW;
        int col = idx - row * HW;
        *(float4*)&tile[row * STRIDE + col] = g4[i];
    }
    __syncthreads();

    // ---- phase 2a: argmax, 16 threads per map (strided, conflict-free) ----
    const int map = tid >> 4;
    const int sub = tid & 15;
    float mv = -3.4e38f;
    int mi = 0;
#ifndef HAVE_WMMA_F32X4
    float s0 = 0.f, s1 = 0.f, s2 = 0.f, s3 = 0.f, s4 = 0.f;
#endif
    for (int p = sub; p < HW; p += 16) {
        float v = tile[map * STRIDE + p];
        if (v > mv) { mv = v; mi = p; }   // ascending p => first-index tie-break
#ifndef HAVE_WMMA_F32X4
        int y = p / 28;
        int x = p - y * 28;
        float fy = (float)y, fx = (float)x;
        s0 += v;
        s1 += v * fy;
        s2 += v * fx;
        s3 += v * fy * fy;
        s4 += v * fx * fx;
#endif
    }
    amaxV[tid] = mv;
    amaxI[tid] = mi;
#ifndef HAVE_WMMA_F32X4
    thrS[tid * 5 + 0] = s0; thrS[tid * 5 + 1] = s1; thrS[tid * 5 + 2] = s2;
    thrS[tid * 5 + 3] = s3; thrS[tid * 5 + 4] = s4;
#endif

#ifdef HAVE_WMMA_F32X4
    // ---- phase 2b: WMMA accumulation, per wave over disjoint K-chunks ----
    {
        const int lane  = tid & 31;
        // wave index as an SGPR so the chunk loop is a scalar branch (EXEC
        // stays all-ones around the wmma, as the ISA requires).
        const int wv    = __builtin_amdgcn_readfirstlane(tid) >> 5;
        const int row   = lane & 15;       // A: map row
        const int khalf = (lane >> 4) * 2; // A/B: K pair base for this half-wave
        const int j     = lane & 15;       // B/D: column

        // one-hot basis coefficients (branchless B build in the loop)
        const float c0 = (j == 0) ? 1.f : 0.f;
        const float c1 = (j == 1) ? 1.f : 0.f;
        const float c2 = (j == 2) ? 1.f : 0.f;
        const float c3 = (j == 3) ? 1.f : 0.f;
        const float c4 = (j == 4) ? 1.f : 0.f;

        // lane's first pixel of its K pair; advances by 32 per iteration
        int p0 = wv * 4 + khalf;
        int y = p0 / 28;
        int x = p0 - y * 28;

        v8f acc = {0.f, 0.f, 0.f, 0.f, 0.f, 0.f, 0.f, 0.f};
        for (int chunk = wv; chunk < 196; chunk += 8) {
            v2f a = *(const v2f*)&tile[row * STRIDE + chunk * 4 + khalf];  // ds_load_b64

            float fy = (float)y;
            float fx = (float)x;
            v2f b;
            b.x = c0 + fy * (c1 + c3 * fy) + fx * (c2 + c4 * fx);
            // neighbor pixel p0+1 (handles x==27 row wrap with selects)
            int wrap = (x == 27) ? 1 : 0;
            float fy1 = fy + (float)wrap;
            float fx1 = wrap ? 0.f : (fx + 1.f);
            b.y = c0 + fy1 * (c1 + c3 * fy1) + fx1 * (c2 + c4 * fx1);

            acc = __builtin_amdgcn_wmma_f32_16x16x4_f32(
                false, a, false, b, (short)0, acc, false, false);

            // advance 32 pixels: x += 4, y += 1, wrap if needed (branchless)
            x += 4;
            y += 1;
            int adj = (x >= 28) ? 1 : 0;
            x -= 28 * adj;
            y += adj;
        }
        if (j < 5) {
            int half = lane >> 4;
#pragma unroll
            for (int v = 0; v < 8; ++v) {
                int r = v + half * 8;  // D row held in accumulator VGPR v
                sumsA[wv * 80 + r * 5 + j] = acc[v];
            }
        }
    }
#endif
    __syncthreads();

    // ---- phase 3: combine sums + argmax per map ----
#ifdef HAVE_WMMA_F32X4
    if (tid < 80) {
        float s = 0.f;
#pragma unroll
        for (int w = 0; w < 8; ++w) s += sumsA[w * 80 + tid];
        sums[tid] = s;
    }
#else
    if (tid < 80) {  // tid -> (m = tid/5, j = tid%5)
        int m = tid / 5, j = tid - (tid / 5) * 5;
        float s = 0.f;
        for (int k = 0; k < 16; ++k) s += thrS[(m * 16 + k) * 5 + j];
        sums[tid] = s;
    }
#endif
    if (sub == 0) {  // one leader per map: reduce 16 candidates, tie -> smaller idx
        float bv = amaxV[tid];
        int bi = amaxI[tid];
        for (int k = 1; k < 16; ++k) {
            float v = amaxV[tid + k];
            int ii = amaxI[tid + k];
            if (v > bv || (v == bv && ii < bi)) { bv = v; bi = ii; }
        }
        mapIdx[map] = bi;
    }
    __syncthreads();

    if (tid < 16) {
        int m = tid;
        int bi = mapIdx[m];
        float iy = (float)(bi / 28);
        float ix = (float)(bi - (bi / 28) * 28);
        float S0  = sums[m * 5 + 0];
        float Sy  = sums[m * 5 + 1];
        float Sx  = sums[m * 5 + 2];
        float Syy = sums[m * 5 + 3];
        float Sxx = sums[m * 5 + 4];
        contrib[m] = (iy * iy + ix * ix) * S0 - 2.f * iy * Sy - 2.f * ix * Sx + Syy + Sxx;
    }
    __syncthreads();
    if (tid == 0) {
        float t = 0.f;
        for (int m = 0; m < 16; ++m) t += contrib[m];
        wsDis[g] = t;
    }
}

// ---------------------------------------------------------------------------
// Kernel B: div_loss partials. block=(128), grid=(7,64): one pixel per thread,
// loop over 512 channels (coalesced plane segments; data is L2-resident after
// kernel A since the whole tensor (102.8MB) fits in the 192MB L2).
// ---------------------------------------------------------------------------
__global__ __launch_bounds__(128) void div_kernel(const float* __restrict__ M,
                                                  float* __restrict__ wsDiv) {
    const int HW = 784, C = 512;
    __shared__ float red[128];
    const int tid = threadIdx.x;
    const int chunkIdx = blockIdx.x;  // 0..6 (112 pixels each)
    const int b = blockIdx.y;         // 0..63

    float contrib = 0.f;
    if (tid < 112) {
        const int pix = chunkIdx * 112 + tid;
        const float* base = M + (size_t)b * C * HW + pix;
        float max1 = -3.4e38f, max2 = -3.4e38f, sum = 0.f;
#pragma unroll 4
        for (int c = 0; c < C; ++c) {
            float v = base[(size_t)c * HW];
            sum += v;
            float nmax2 = fmaxf(max2, fminf(max1, v));  // branchless top-2
            max1 = fmaxf(max1, v);
            max2 = nmax2;
        }
        float m1 = fmaxf(max1, NEG_FILL);
        float m2 = fmaxf(max2, NEG_FILL);
        contrib = m1 * (sum - max1) + m2 * max1;
    }
    red[tid] = contrib;
    __syncthreads();
    for (int s = 64; s > 0; s >>= 1) {
        if (tid < s) red[tid] += red[tid + s];
        __syncthreads();
    }
    if (tid == 0) wsDiv[b * 7 + chunkIdx] = red[0];
}

// ---------------------------------------------------------------------------
// Kernel C: deterministic final reduction of partials -> (dis_loss, div_loss)
// ---------------------------------------------------------------------------
__global__ __launch_bounds__(256) void finalize_kernel(const float* __restrict__ wsDis,
                                                       const float* __restrict__ wsDiv,
                                                       float* __restrict__ out) {
    __shared__ float red[256];
    const int tid = threadIdx.x;
    const float invN = 1.0f / 25690112.0f;  // 64*512*28*28

    float s = 0.f;
    for (int i = tid; i < 2048; i += 256) s += wsDis[i];
    red[tid] = s;
    __syncthreads();
    for (int st = 128; st > 0; st >>= 1) {
        if (tid < st) red[tid] += red[tid + st];
        __syncthreads();
    }
    if (tid == 0) out[0] = red[0] * invN;
    __syncthreads();

    float s2 = 0.f;
    for (int i = tid; i < 448; i += 256) s2 += wsDiv[i];
    red[tid] = s2;
    __syncthreads();
    for (int st = 128; st > 0; st >>= 1) {
        if (tid < st) red[tid] += red[tid + st];
        __syncthreads();
    }
    if (tid == 0) out[1] = red[0] * invN;
}

extern "C" void kernel_launch(void* const* d_in, const int* in_sizes, int n_in,
                              void* d_out, int out_size, void* d_ws, size_t ws_size,
                              hipStream_t stream) {
    const float* M = (const float*)d_in[0];
    float* out = (float*)d_out;
    float* ws = (float*)d_ws;
    float* wsDis = ws;         // 2048 partials (one per 16-map block)
    float* wsDiv = ws + 2048;  // 448 partials (64 b x 7 pixel-chunks)

    dis_kernel<<<2048, 256, 0, stream>>>(M, wsDis);
    div_kernel<<<dim3(7, 64), 128, 0, stream>>>(M, wsDiv);
    finalize_kernel<<<1, 256, 0, stream>>>(wsDis, wsDiv, out);
}